// ElmanRNN_pytorch_module_v2_10728828305911
// MI455X (gfx1250) — compile-verified
//
#include <hip/hip_runtime.h>
#include <math.h>

// Problem dims (match reference)
#define BQ 32
#define TQ 512
#define IQ 512
#define HQ 1024
#define OQ 512

#define RNN_BLOCKS 32           // persistent blocks for the scan (all resident)
// TDM pads 2 dwords every 256 dwords -> row stride 1032 dwords, float2-aligned,
// lane bank index n*8 mod 64 (2-way conflict worst case).
#define LDS_ROW_STRIDE 1032

typedef __attribute__((ext_vector_type(2))) float v2f;
typedef __attribute__((ext_vector_type(8))) float v8f;
typedef __attribute__((ext_vector_type(4))) unsigned v4u;
typedef __attribute__((ext_vector_type(8))) unsigned v8u;

// ---------------------------------------------------------------------------
// Per-lane fragment mapping for V_WMMA_F32_16X16X4_F32 (wave32), ISA §7.12.2:
//   A (16x4):  lane l -> row m = l&15 ; VGPR j holds K = 2*(l>>4) + j
//   B (4x16):  lane l -> col n = l&15 ; VGPR j holds K = 2*(l>>4) + j
//   C/D(16x16):lane l, VGPR r -> (M = r + 8*(l>>4), N = l&15)
// ---------------------------------------------------------------------------

// Generic fp32 WMMA GEMM: out[M,N] = A[M,K] * Bm[N,K]^T + b1[n] (+ b2[n])
// One wave per 16x16 tile, 8 waves per block. K % 8 == 0.
__global__ __launch_bounds__(256) void k_gemm_bias(
    const float* __restrict__ A, const float* __restrict__ Bm,
    const float* __restrict__ b1, const float* __restrict__ b2,
    float* __restrict__ out, int K, int ntiles, int ldc) {
  const int lane = threadIdx.x & 31;
  const int wave = threadIdx.x >> 5;
  const int tile = blockIdx.x * 8 + wave;
  const int mt = tile / ntiles;
  const int nt = tile - mt * ntiles;
  const int khalf = (lane >> 4) << 1;                 // 0 or 2
  const float* arow = A  + (size_t)((mt << 4) + (lane & 15)) * K;
  const float* brow = Bm + (size_t)((nt << 4) + (lane & 15)) * K;

  v8f c0 = {}; v8f c1 = {};
  for (int k = 0; k < K; k += 8) {
    if ((k & 255) == 0) {                             // global_prefetch_b8
      __builtin_prefetch(arow + k + 512, 0, 3);
      __builtin_prefetch(brow + k + 512, 0, 3);
    }
    const float2 av0 = *(const float2*)(arow + k + khalf);
    const float2 bv0 = *(const float2*)(brow + k + khalf);
    const float2 av1 = *(const float2*)(arow + k + 4 + khalf);
    const float2 bv1 = *(const float2*)(brow + k + 4 + khalf);
    v2f a0 = {av0.x, av0.y}, b0 = {bv0.x, bv0.y};
    v2f a1 = {av1.x, av1.y}, b1v = {bv1.x, bv1.y};
    c0 = __builtin_amdgcn_wmma_f32_16x16x4_f32(false, a0, false, b0,  (short)0, c0, false, false);
    c1 = __builtin_amdgcn_wmma_f32_16x16x4_f32(false, a1, false, b1v, (short)0, c1, false, false);
  }

  const int n_g   = (nt << 4) + (lane & 15);
  float bias = b1[n_g];
  if (b2) bias += b2[n_g];
  const int Mbase = (mt << 4) + ((lane >> 4) << 3);
#pragma unroll
  for (int r = 0; r < 8; ++r)
    out[(size_t)(Mbase + r) * ldc + n_g] = c0[r] + c1[r] + bias;
}

// LDS dword index of W_hh element (row n, col k) under TDM padding:
// 2 extra dwords inserted after every 256 dwords streamed.
__device__ __forceinline__ int lds_idx(int n, int k) {
  return n * LDS_ROW_STRIDE + k + ((k >> 8) << 1);
}

// ---------------------------------------------------------------------------
// Persistent recurrent scan: 512 steps of h = tanh(pre[:,t,:] + h @ W_hh^T)
// 32 blocks x 128 threads (4 waves). Block owns 2 n-tiles (32 rows of W_hh),
// staged once into LDS by the Tensor Data Mover; wave -> (mt = w&1, ntl = w>>1).
// Grid-wide sense barrier between steps (h ping-pong in global ws).
// ---------------------------------------------------------------------------
__global__ __launch_bounds__(128) void k_rnn(
    const float* __restrict__ pre, const float* __restrict__ Whh,
    float* __restrict__ hA, float* __restrict__ hB,
    float* __restrict__ z, int* __restrict__ bar) {
  __shared__ float ldsB[32 * LDS_ROW_STRIDE];         // 132,096 B of the 320 KB WGP LDS

  const int tid  = threadIdx.x;
  const int lane = tid & 31;
  const int wave = tid >> 5;                          // 0..3
  const int mt   = wave & 1;                          // 2 m-tiles (B=32 rows)
  const int ntl  = wave >> 1;                         // 0..1 local n-tile

  // ---- TDM: DMA this block's 32 W_hh rows (32x1024 f32 tile) into LDS ----
  if (wave == 0) {
    const unsigned long long gaddr =
        (unsigned long long)(const void*)(Whh + (size_t)blockIdx.x * 32 * HQ);
    const unsigned ldsaddr = (unsigned)(size_t)(const void*)&ldsB[0];
    // D# group0: count=1 | lds_addr | global_addr[56:0] | type=2
    const v4u g0 = { 1u, ldsaddr, (unsigned)gaddr,
                     (unsigned)((gaddr >> 32) & 0x01FFFFFFu) | 0x80000000u };
    // D# group1: data_size=4B, pad_enable, pad_interval=256dw, pad_amount=2dw,
    //            tensor_dim0=1024, tensor_dim1=32, tile 1024x32, dim0_stride=1024
    const v8u g1 = { 0x03D20000u,          // mask=0 | ds=2<<16 | pe<<20 | pi=7<<22 | pa=1<<25
                     0x04000000u,          // tensor_dim0 low16 in [31:16]
                     (32u << 16),          // tensor_dim1 low16
                     (1024u << 16),        // tile_dim0
                     32u,                  // tile_dim1
                     1024u,                // tensor_dim0_stride low32
                     0u, 0u };
    const v4u gz = { 0u, 0u, 0u, 0u };     // groups 2/3: dims unused (2D tile)
    asm volatile("tensor_load_to_lds %0, %1, %2, %3"
                 :: "s"(g0), "s"(g1), "s"(gz), "s"(gz) : "memory");
    __builtin_amdgcn_s_wait_tensorcnt(0);
  }
  __syncthreads();

  const int khalf = (lane >> 4) << 1;
  const int mrow  = (mt << 4) + (lane & 15);          // A row = h row
  const int brow  = (ntl << 4) + (lane & 15);         // LDS B row (0..31)
  const int nt    = blockIdx.x * 2 + ntl;
  const int n_g   = (nt << 4) + (lane & 15);
  const int Mbase = (mt << 4) + ((lane >> 4) << 3);

  const float* hprev = hA;
  float*       hnext = hB;
  for (int t = 0; t < TQ; ++t) {
    const float* arow = hprev + (size_t)mrow * HQ;
    v8f c0 = {}; v8f c1 = {};
#pragma unroll 4
    for (int k = 0; k < HQ; k += 8) {
      const float2 av0 = *(const float2*)(arow + k + khalf);
      const float2 av1 = *(const float2*)(arow + k + 4 + khalf);
      const float2 bv0 = *(const float2*)(&ldsB[lds_idx(brow, k + khalf)]);
      const float2 bv1 = *(const float2*)(&ldsB[lds_idx(brow, k + 4 + khalf)]);
      v2f a0 = {av0.x, av0.y}, b0 = {bv0.x, bv0.y};
      v2f a1 = {av1.x, av1.y}, b1v = {bv1.x, bv1.y};
      c0 = __builtin_amdgcn_wmma_f32_16x16x4_f32(false, a0, false, b0,  (short)0, c0, false, false);
      c1 = __builtin_amdgcn_wmma_f32_16x16x4_f32(false, a1, false, b1v, (short)0, c1, false, false);
    }
#pragma unroll
    for (int r = 0; r < 8; ++r) {
      const int m = Mbase + r;                        // batch index
      const float hv = tanhf(c0[r] + c1[r] +
                             pre[(size_t)m * (TQ * HQ) + (size_t)t * HQ + n_g]);
      hnext[m * HQ + n_g] = hv;
      z[(size_t)m * (TQ * HQ) + (size_t)t * HQ + n_g] = hv;
    }

    // ---- grid barrier (release h writes, acquire before next step) ----
    __threadfence();
    __syncthreads();
    if (tid == 0) {
      const int arrived = atomicAdd(&bar[0], 1);
      if (arrived == RNN_BLOCKS - 1) {
        __hip_atomic_store(&bar[0], 0, __ATOMIC_RELAXED, __HIP_MEMORY_SCOPE_AGENT);
        __threadfence();
        atomicAdd(&bar[1], 1);                        // generation++ : release step t
      } else {
        while (__hip_atomic_load(&bar[1], __ATOMIC_ACQUIRE, __HIP_MEMORY_SCOPE_AGENT) <= t)
          __builtin_amdgcn_s_sleep(2);
      }
    }
    __syncthreads();

    const float* tmp = hprev; hprev = hnext; hnext = (float*)tmp;
  }
}

// Row softmax over O=512 columns; one block per row.
__global__ __launch_bounds__(256) void k_softmax(
    const float* __restrict__ logits, float* __restrict__ out) {
  __shared__ float red[256];
  const int row = blockIdx.x;
  const int tid = threadIdx.x;
  const float* lr = logits + (size_t)row * OQ;
  const float v0 = lr[tid], v1 = lr[tid + 256];
  red[tid] = fmaxf(v0, v1);
  __syncthreads();
  for (int s = 128; s > 0; s >>= 1) {
    if (tid < s) red[tid] = fmaxf(red[tid], red[tid + s]);
    __syncthreads();
  }
  const float mx = red[0];
  __syncthreads();
  const float e0 = expf(v0 - mx), e1 = expf(v1 - mx);
  red[tid] = e0 + e1;
  __syncthreads();
  for (int s = 128; s > 0; s >>= 1) {
    if (tid < s) red[tid] += red[tid + s];
    __syncthreads();
  }
  const float inv = 1.0f / red[0];
  out[(size_t)row * OQ + tid]       = e0 * inv;
  out[(size_t)row * OQ + tid + 256] = e1 * inv;
}

// Per-launch init: copy h0 -> h ping buffer, zero barrier state (graph-safe).
__global__ void k_init(const float* __restrict__ h0, float* __restrict__ hA,
                       int* __restrict__ bar) {
  const int i = blockIdx.x * blockDim.x + threadIdx.x;
  if (i < BQ * HQ) hA[i] = h0[i];
  if (i < 2) bar[i] = 0;
}

extern "C" void kernel_launch(void* const* d_in, const int* in_sizes, int n_in,
                              void* d_out, int out_size, void* d_ws, size_t ws_size,
                              hipStream_t stream) {
  (void)in_sizes; (void)n_in; (void)out_size; (void)ws_size;
  const float* x    = (const float*)d_in[0];   // [B,T,I]
  const float* h0   = (const float*)d_in[1];   // [1,B,H]
  const float* Wih  = (const float*)d_in[2];   // [H,I]
  const float* Whh  = (const float*)d_in[3];   // [H,H]
  const float* bih  = (const float*)d_in[4];   // [H]
  const float* bhh  = (const float*)d_in[5];   // [H]
  const float* Wout = (const float*)d_in[6];   // [O,H]
  const float* bout = (const float*)d_in[7];   // [O]

  float* out = (float*)d_out;                          // [B,T,O]
  float* z   = out + (size_t)BQ * TQ * OQ;             // [B,T,H]

  char*  ws   = (char*)d_ws;
  float* pre  = (float*)ws;                            // 64 MB: [B,T,H] (both biases folded)
  float* hA   = (float*)(ws + (size_t)BQ * TQ * HQ * 4);
  float* hB   = hA + BQ * HQ;
  int*   bar  = (int*)(hB + BQ * HQ);
  float* logits = pre;                                 // reuse pre region after the scan

  k_init<<<(BQ * HQ + 255) / 256, 256, 0, stream>>>(h0, hA, bar);
  // pre = x @ Wih^T + bih + bhh : M=16384, N=1024 (64 n-tiles), K=512
  k_gemm_bias<<<(1024 * 64) / 8, 256, 0, stream>>>(x, Wih, bih, bhh, pre, IQ, 64, HQ);
  // 512 sequential steps, persistent grid
  k_rnn<<<RNN_BLOCKS, 128, 0, stream>>>(pre, Whh, hA, hB, z, bar);
  // logits = z @ Wout^T + bout : M=16384, N=512 (32 n-tiles), K=1024
  k_gemm_bias<<<(1024 * 32) / 8, 256, 0, stream>>>(z, Wout, bout, nullptr, logits, HQ, 32, OQ);
  k_softmax<<<BQ * TQ, 256, 0, stream>>>(logits, out);
}